// RandomElasticDeformation3D_20839181320474
// MI455X (gfx1250) — compile-verified
//
#include <hip/hip_runtime.h>
#include <math.h>

#define ALPHA 35.0f
#define SIGMA 2.5f
#define B_ 2
#define D_ 128
#define H_ 160
#define W_ 160
#define C_ 4

typedef __attribute__((ext_vector_type(2))) float v2f;
typedef __attribute__((ext_vector_type(4))) float v4f;
typedef __attribute__((ext_vector_type(8))) float v8f;

// ---------------------------------------------------------------------------
// Setup: fused (Gaussian-SAME-conv ∘ bicubic-upsample) per-axis tables.
// tabs[0..511]    = A_D (128 x 4), ALPHA folded in
// tabs[512..1151] = A_H (160 x 4), shared for the W axis (H_ == W_)
// ---------------------------------------------------------------------------
__device__ __forceinline__ float keys_cubic(float t) {
    const float a = -0.5f;
    t = fabsf(t);
    if (t <= 1.0f) return ((a + 2.0f) * t - (a + 3.0f)) * t * t + 1.0f;
    if (t < 2.0f)  return ((((t - 5.0f) * t) + 8.0f) * t - 4.0f) * a;
    return 0.0f;
}

__global__ void build_tables_kernel(float* __restrict__ tabs) {
    float kg[16];
    float ks = 0.0f;
    for (int t = 0; t < 16; ++t) {
        float ax = (float)(t - 7);
        kg[t] = expf(-(ax * ax) / (2.0f * SIGMA * SIGMA));
        ks += kg[t];
    }
    for (int t = 0; t < 16; ++t) kg[t] /= ks;

    for (int idx = threadIdx.x; idx < 128 + 160; idx += blockDim.x) {
        const int   isD = (idx < 128);
        const int   N   = isD ? D_ : H_;
        const int   n   = isD ? idx : idx - 128;
        float*      dst = isD ? (tabs + n * 4) : (tabs + 512 + n * 4);
        const float scl = isD ? ALPHA : 1.0f;
        float acc[4] = {0.f, 0.f, 0.f, 0.f};
        for (int t = 0; t < 16; ++t) {
            int m = n - 7 + t;
            if (m < 0 || m >= N) continue;               // SAME zero padding
            float xin = (m + 0.5f) * (4.0f / (float)N) - 0.5f;
            float w[4], wsum = 0.0f;
            for (int i = 0; i < 4; ++i) { w[i] = keys_cubic((float)i - xin); wsum += w[i]; }
            float inv = (wsum != 0.0f) ? (1.0f / wsum) : 0.0f;
            for (int i = 0; i < 4; ++i) acc[i] += kg[t] * w[i] * inv;
        }
        for (int i = 0; i < 4; ++i) dst[i] = acc[i] * scl;
    }
}

// ---------------------------------------------------------------------------
// Main kernel: one wave per (b,d,h)-row-pair, WMMA flow + branchless gather.
// ---------------------------------------------------------------------------
__global__ __launch_bounds__(256) void elastic_warp_kernel(
        const float* __restrict__ img,     // (B,D,H,W,C)
        const float* __restrict__ lbl,     // (B,D,H,W,1)
        const float* __restrict__ coarse,  // (B,4,4,4,3)
        const float* __restrict__ tabs,
        float* __restrict__ out) {
    const float* Ad = tabs;          // 128x4 (ALPHA folded)
    const float* Ah = tabs + 512;    // 160x4 (H and W axes)

    const int lane = threadIdx.x & 31;
    const int wave = blockIdx.x * (blockDim.x >> 5) + (threadIdx.x >> 5);
    const int row0 = wave * 2;       // rows (row0, row0+1); row = (b*D+d)*H + h

    const int m  = lane & 15;        // WMMA M (A) / N (B,D) index
    const int k0 = (lane >> 4) * 2;  // K pair held by this lane half

    // ---- A operand: C2^T (rows 0-2 = row0's components, rows 8-10 = row1's)
    v2f a; a.x = 0.0f; a.y = 0.0f;
    {
        int crow = -1, comp = 0;
        if (m < 3)                 { crow = row0;     comp = m;     }
        else if (m >= 8 && m < 11) { crow = row0 + 1; comp = m - 8; }
        if (crow >= 0) {
            const int cb  = crow / (D_ * H_);
            const int rem = crow % (D_ * H_);
            const int cd  = rem / H_;
            const int ch  = rem % H_;
            const float* Adr = Ad + cd * 4;
            const float* Ahr = Ah + ch * 4;
            float s0 = 0.0f, s1 = 0.0f;
            #pragma unroll
            for (int i = 0; i < 4; ++i) {
                const float wd = Adr[i];
                #pragma unroll
                for (int j = 0; j < 4; ++j) {
                    const float wdh = wd * Ahr[j];
                    const float* cp = coarse + (((cb * 4 + i) * 4 + j) * 4) * 3;
                    s0 += wdh * cp[k0 * 3 + comp];
                    s1 += wdh * cp[(k0 + 1) * 3 + comp];
                }
            }
            a.x = s0; a.y = s1;
        }
    }

    // ---- this lane's gather voxel position
    const int rowg = row0 + (lane >> 4);
    const int gb   = rowg / (D_ * H_);
    const int remg = rowg % (D_ * H_);
    const int gd   = remg / H_;
    const int gh   = remg % H_;

    const float* ib   = img + gb * (D_ * H_ * W_ * C_);   // per-batch image base
    const float* lb   = lbl + gb * (D_ * H_ * W_);        // per-batch label base
    const int rowBase = (gd * H_ + gh) * W_;              // within batch, voxel row
    float* outImg = out + (gb * (D_ * H_ * W_) + rowBase) * C_;
    float* outLbl = out + (B_ * D_ * H_ * W_ * C_) + gb * (D_ * H_ * W_) + rowBase;

    // ---- software-pipelined B operand (A_W^T columns); k0 even -> 8B aligned
    v2f bm = *reinterpret_cast<const v2f*>(Ah + m * 4 + k0);

    #pragma unroll 2
    for (int w0 = 0; w0 < W_; w0 += 16) {
        const int wv = w0 + m;

        v8f cz = {};
        v8f dacc = __builtin_amdgcn_wmma_f32_16x16x4_f32(
            false, a, false, bm, (short)0, cz, false, false);

        // unconditional clamped prefetch of next tile's B columns (value dead
        // on the last iteration) -> no branch, schedulable with gather loads
        {
            const int pw = min(wv + 16, W_ - 1);
            bm = *reinterpret_cast<const v2f*>(Ah + pw * 4 + k0);
        }

        const float fd = dacc[0];   // rows 0-2 (lanes 0-15) / rows 8-10 (lanes 16-31)
        const float fh = dacc[1];
        const float fw = dacc[2];

        const float zd = (float)gd + fd;
        const float zh = (float)gh + fh;
        const float zw = (float)wv + fw;

        // ---- branchless trilinear: clamp indices, zero OOB axis weights ----
        const float fld = floorf(zd), flh = floorf(zh), flw = floorf(zw);
        const int   i0 = (int)fld, j0 = (int)flh, q0 = (int)flw;
        const int   i1 = i0 + 1,   j1 = j0 + 1,   q1 = q0 + 1;
        const float td = zd - fld, th = zh - flh, tw = zw - flw;

        const float wd0 = ((unsigned)i0 < (unsigned)D_) ? (1.0f - td) : 0.0f;
        const float wd1 = ((unsigned)i1 < (unsigned)D_) ? td          : 0.0f;
        const float wh0 = ((unsigned)j0 < (unsigned)H_) ? (1.0f - th) : 0.0f;
        const float wh1 = ((unsigned)j1 < (unsigned)H_) ? th          : 0.0f;
        const float ww0 = ((unsigned)q0 < (unsigned)W_) ? (1.0f - tw) : 0.0f;
        const float ww1 = ((unsigned)q1 < (unsigned)W_) ? tw          : 0.0f;

        const int i0c = min(max(i0, 0), D_ - 1), i1c = min(max(i1, 0), D_ - 1);
        const int j0c = min(max(j0, 0), H_ - 1), j1c = min(max(j1, 0), H_ - 1);
        const int q0c = min(max(q0, 0), W_ - 1), q1c = min(max(q1, 0), W_ - 1);

        const int r00 = (i0c * H_ + j0c) * W_;
        const int r01 = (i0c * H_ + j1c) * W_;
        const int r10 = (i1c * H_ + j0c) * W_;
        const int r11 = (i1c * H_ + j1c) * W_;

        // 8 independent 128-bit loads (always in-bounds) -> full MLP
        const v4f v000 = *reinterpret_cast<const v4f*>(ib + (r00 + q0c) * C_);
        const v4f v001 = *reinterpret_cast<const v4f*>(ib + (r00 + q1c) * C_);
        const v4f v010 = *reinterpret_cast<const v4f*>(ib + (r01 + q0c) * C_);
        const v4f v011 = *reinterpret_cast<const v4f*>(ib + (r01 + q1c) * C_);
        const v4f v100 = *reinterpret_cast<const v4f*>(ib + (r10 + q0c) * C_);
        const v4f v101 = *reinterpret_cast<const v4f*>(ib + (r10 + q1c) * C_);
        const v4f v110 = *reinterpret_cast<const v4f*>(ib + (r11 + q0c) * C_);
        const v4f v111 = *reinterpret_cast<const v4f*>(ib + (r11 + q1c) * C_);

        // ---- nearest-neighbor label (round-half-even), clamped + masked ----
        const int ni = (int)rintf(zd);
        const int nj = (int)rintf(zh);
        const int nq = (int)rintf(zw);
        const bool lvalid = ((unsigned)ni < (unsigned)D_) &&
                            ((unsigned)nj < (unsigned)H_) &&
                            ((unsigned)nq < (unsigned)W_);
        const int nic = min(max(ni, 0), D_ - 1);
        const int njc = min(max(nj, 0), H_ - 1);
        const int nqc = min(max(nq, 0), W_ - 1);
        float lv = lb[(nic * H_ + njc) * W_ + nqc];
        lv = lvalid ? lv : 0.0f;

        const float w000 = wd0 * wh0 * ww0, w001 = wd0 * wh0 * ww1;
        const float w010 = wd0 * wh1 * ww0, w011 = wd0 * wh1 * ww1;
        const float w100 = wd1 * wh0 * ww0, w101 = wd1 * wh0 * ww1;
        const float w110 = wd1 * wh1 * ww0, w111 = wd1 * wh1 * ww1;

        v4f ov = w000 * v000 + w001 * v001 + w010 * v010 + w011 * v011
               + w100 * v100 + w101 * v101 + w110 * v110 + w111 * v111;

        // non-temporal stores: keep the 131 MB gather working set resident in L2
        __builtin_nontemporal_store(ov, reinterpret_cast<v4f*>(outImg + wv * C_));
        __builtin_nontemporal_store(lv, outLbl + wv);
    }
}

extern "C" void kernel_launch(void* const* d_in, const int* in_sizes, int n_in,
                              void* d_out, int out_size, void* d_ws, size_t ws_size,
                              hipStream_t stream) {
    const float* img    = (const float*)d_in[0];   // (2,128,160,160,4)
    const float* lbl    = (const float*)d_in[1];   // (2,128,160,160,1)
    const float* coarse = (const float*)d_in[2];   // (2,4,4,4,3)
    float* tabs = (float*)d_ws;                    // 1152 floats
    float* out  = (float*)d_out;

    build_tables_kernel<<<1, 256, 0, stream>>>(tabs);

    const int rowPairs      = B_ * D_ * H_ / 2;          // 20480
    const int wavesPerBlock = 256 / 32;                  // 8
    const int blocks        = rowPairs / wavesPerBlock;  // 2560, exact
    elastic_warp_kernel<<<blocks, 256, 0, stream>>>(img, lbl, coarse, tabs, out);
}